// GCNClassifier_78915729097080
// MI455X (gfx1250) — compile-verified
//
#include <hip/hip_runtime.h>

// Problem constants (match reference)
#define NN   100000
#define EE   1600000
#define DD   128
#define LL   3
#define GG   128
#define CC   10
#define HH   64
#define EPSV 1e-5f

typedef float v2f __attribute__((ext_vector_type(2)));
typedef float v8f __attribute__((ext_vector_type(8)));

__device__ __forceinline__ void atomAddF(float* p, float v) {
    unsafeAtomicAdd(p, v);   // hardware global_atomic_add_f32
}

__device__ __forceinline__ v8f wmma_f32_k4(v2f a, v2f b, v8f c) {
    return __builtin_amdgcn_wmma_f32_16x16x4_f32(
        /*neg_a=*/false, a, /*neg_b=*/false, b,
        /*c_mod=*/(short)0, c, /*reuse_a=*/false, /*reuse_b=*/false);
}

// ---------------- degree / norm precompute ----------------

__global__ void k_fill(float* __restrict__ p, float val, int n) {
    int i = blockIdx.x * blockDim.x + threadIdx.x;
    if (i < n) p[i] = val;
}

__global__ void k_deg_edges(const int* __restrict__ dst, float* __restrict__ deg, int e) {
    int i = blockIdx.x * blockDim.x + threadIdx.x;
    if (i < e) atomAddF(&deg[dst[i]], 1.0f);
}

__global__ void k_dinv(const float* __restrict__ deg, float* __restrict__ dinv,
                       float* __restrict__ selfnorm, int n) {
    int i = blockIdx.x * blockDim.x + threadIdx.x;
    if (i >= n) return;
    float dg = deg[i];            // >= 1 (self loop)
    dinv[i] = rsqrtf(dg);
    selfnorm[i] = 1.0f / dg;      // dinv^2 for the self loop edge
}

__global__ void k_norm(const int* __restrict__ src, const int* __restrict__ dst,
                       const float* __restrict__ dinv, float* __restrict__ norm, int e) {
    int i = blockIdx.x * blockDim.x + threadIdx.x;
    if (i < e) norm[i] = dinv[src[i]] * dinv[dst[i]];
}

// ---------------- WMMA f32 GEMM: out[N,128] = h[N,128] @ W[128,128] ----------------
// One wave computes a 16x64 output strip (four 16x16 col tiles) so the A fragment
// loaded each k-step feeds 4 WMMAs. Block = 8 waves = 4 row tiles x 2 col halves.
// A layout (ISA 7.12.2, 32-bit A 16x4): lanes 0-15 -> K={0,1}, lanes 16-31 -> K={2,3}.
// B layout mirrors (K striped across lane halves, N = lane%16).
// C/D layout: VGPR r -> M = r + 8*(lane/16), N = lane%16.

__global__ __launch_bounds__(256) void k_gemm(const float* __restrict__ h,
                                              const float* __restrict__ W,
                                              float* __restrict__ out) {
    const int wave = threadIdx.x >> 5;       // 0..7
    const int lane = threadIdx.x & 31;
    const int half = lane >> 4;              // 0: K pair {0,1}; 1: K pair {2,3}
    const int lrow = lane & 15;

    const int rowTile = blockIdx.x * 4 + (wave >> 1);   // 16-row tile index
    if (rowTile >= NN / 16) return;                     // wave-uniform guard
    const int rowBase = rowTile << 4;
    const int colBase = (wave & 1) << 6;                // 0 or 64

    const float* arow = h + (size_t)(rowBase + lrow) * DD;
    const float* bcol = W + colBase + lrow;

    v8f c0 = {0.f, 0.f, 0.f, 0.f, 0.f, 0.f, 0.f, 0.f};
    v8f c1 = c0, c2 = c0, c3 = c0;

#pragma unroll 4
    for (int kk = 0; kk < 32; ++kk) {
        const int k0 = (kk << 2) + (half << 1);
        const float2 av = *reinterpret_cast<const float2*>(arow + k0);  // b64 load
        v2f a; a.x = av.x; a.y = av.y;
        const float* bk0 = bcol + (size_t)k0 * DD;
        const float* bk1 = bcol + (size_t)(k0 + 1) * DD;
        v2f b0; b0.x = bk0[0];  b0.y = bk1[0];
        v2f b1; b1.x = bk0[16]; b1.y = bk1[16];
        v2f b2; b2.x = bk0[32]; b2.y = bk1[32];
        v2f b3; b3.x = bk0[48]; b3.y = bk1[48];
        c0 = wmma_f32_k4(a, b0, c0);
        c1 = wmma_f32_k4(a, b1, c1);
        c2 = wmma_f32_k4(a, b2, c2);
        c3 = wmma_f32_k4(a, b3, c3);
    }

    float* orow = out + (size_t)rowBase * DD + colBase + lrow;
#pragma unroll
    for (int r = 0; r < 8; ++r) {
        const size_t off = (size_t)(r + half * 8) * DD;
        orow[off]      = c0[r];
        orow[off + 16] = c1[r];
        orow[off + 32] = c2[r];
        orow[off + 48] = c3[r];
    }
}

// ---------------- self-loop init: agg = hw * selfnorm[node] ----------------

__global__ __launch_bounds__(256) void k_selfinit(const float4* __restrict__ hw4,
                                                  const float* __restrict__ selfnorm,
                                                  float4* __restrict__ agg4, int n4) {
    int i = blockIdx.x * blockDim.x + threadIdx.x;
    if (i >= n4) return;
    int node = i >> 5;               // 32 float4 per 128-float row
    float s = selfnorm[node];
    float4 v = hw4[i];
    v.x *= s; v.y *= s; v.z *= s; v.w *= s;
    agg4[i] = v;
}

// ---------------- edge scatter: agg[dst] += hw[src] * norm ----------------
// One wave per edge: one b128 gather per lane + 4 f32 atomics (wave covers 512B row).

__global__ __launch_bounds__(256) void k_scatter(const float* __restrict__ hw,
                                                 const int* __restrict__ src,
                                                 const int* __restrict__ dst,
                                                 const float* __restrict__ norm,
                                                 float* __restrict__ agg, int e) {
    int w = (blockIdx.x * blockDim.x + threadIdx.x) >> 5;
    if (w >= e) return;
    int lane = threadIdx.x & 31;
    int s = src[w], t = dst[w];
    float nr = norm[w];
    const float4 hv = *reinterpret_cast<const float4*>(hw + (size_t)s * DD + lane * 4);
    float* ap = agg + (size_t)t * DD + lane * 4;
    atomAddF(ap + 0, hv.x * nr);
    atomAddF(ap + 1, hv.y * nr);
    atomAddF(ap + 2, hv.z * nr);
    atomAddF(ap + 3, hv.w * nr);
}

// ---------------- bias + BatchNorm(eval) + ReLU, in place ----------------

__global__ __launch_bounds__(256) void k_bn_relu(float* __restrict__ h,
                                                 const float* __restrict__ b,
                                                 const float* __restrict__ gamma,
                                                 const float* __restrict__ beta,
                                                 const float* __restrict__ mean,
                                                 const float* __restrict__ var, int total) {
    int i = blockIdx.x * blockDim.x + threadIdx.x;
    if (i >= total) return;
    int d = i & (DD - 1);
    float sc = gamma[d] * rsqrtf(var[d] + EPSV);
    float v = (h[i] + b[d] - mean[d]) * sc + beta[d];
    h[i] = v > 0.f ? v : 0.f;
}

// ---------------- pooling ----------------

__global__ void k_counts(const int* __restrict__ batch, float* __restrict__ counts, int n) {
    int i = blockIdx.x * blockDim.x + threadIdx.x;
    if (i < n) atomAddF(&counts[batch[i]], 1.0f);
}

__global__ __launch_bounds__(256) void k_poolsum(const float* __restrict__ h,
                                                 const int* __restrict__ batch,
                                                 float* __restrict__ pooled, int n) {
    int w = (blockIdx.x * blockDim.x + threadIdx.x) >> 5;
    if (w >= n) return;
    int lane = threadIdx.x & 31;
    int g = batch[w];
    const float4 hv = *reinterpret_cast<const float4*>(h + (size_t)w * DD + lane * 4);
    float* pp = pooled + (size_t)g * DD + lane * 4;
    atomAddF(pp + 0, hv.x);
    atomAddF(pp + 1, hv.y);
    atomAddF(pp + 2, hv.z);
    atomAddF(pp + 3, hv.w);
}

// ---------------- MLP head ----------------
// z = relu((pooled_sum @ fc1)/count + b1)   (division distributes over the dot)

__global__ void k_fc1(const float* __restrict__ pooled, const float* __restrict__ counts,
                      const float* __restrict__ w1, const float* __restrict__ b1,
                      float* __restrict__ z) {
    int i = blockIdx.x * blockDim.x + threadIdx.x;
    if (i >= GG * HH) return;
    int g = i / HH, j = i % HH;
    const float* pr = pooled + (size_t)g * DD;
    float acc = 0.f;
#pragma unroll 8
    for (int d = 0; d < DD; ++d) acc += pr[d] * w1[d * HH + j];
    float cnt = counts[g];
    if (cnt < 1.f) cnt = 1.f;
    acc = acc / cnt + b1[j];
    z[i] = acc > 0.f ? acc : 0.f;
}

__global__ void k_fc2(const float* __restrict__ z, const float* __restrict__ w2,
                      const float* __restrict__ b2, float* __restrict__ out) {
    int i = blockIdx.x * blockDim.x + threadIdx.x;
    if (i >= GG * CC) return;
    int g = i / CC, cc = i % CC;
    const float* zr = z + (size_t)g * HH;
    float acc = b2[cc];
#pragma unroll 8
    for (int j = 0; j < HH; ++j) acc += zr[j] * w2[j * CC + cc];
    out[i] = acc;
}

// ---------------- host launcher ----------------

static inline int cdiv(long long a, int b) { return (int)((a + b - 1) / b); }

extern "C" void kernel_launch(void* const* d_in, const int* in_sizes, int n_in,
                              void* d_out, int out_size, void* d_ws, size_t ws_size,
                              hipStream_t stream) {
    const float* x     = (const float*)d_in[0];
    const int*   eidx  = (const int*)d_in[1];
    const int*   src   = eidx;
    const int*   dst   = eidx + EE;
    const int*   batch = (const int*)d_in[2];
    const float* Ws    = (const float*)d_in[3];
    const float* bs    = (const float*)d_in[4];
    const float* gam   = (const float*)d_in[5];
    const float* bet   = (const float*)d_in[6];
    const float* mu    = (const float*)d_in[7];
    const float* var   = (const float*)d_in[8];
    const float* fc1w  = (const float*)d_in[9];
    const float* fc1b  = (const float*)d_in[10];
    const float* fc2w  = (const float*)d_in[11];
    const float* fc2b  = (const float*)d_in[12];
    float* out = (float*)d_out;

    // workspace layout (floats); ~110 MB total
    float* ws       = (float*)d_ws;
    float* deg      = ws;                          // N
    float* dinv     = deg + NN;                    // N
    float* selfn    = dinv + NN;                   // N
    float* norm     = selfn + NN;                  // E
    float* bufA     = norm + EE;                   // N*D (agg / h, in-place BN)
    float* bufB     = bufA + (size_t)NN * DD;      // N*D (hw = h @ W)
    float* pooled   = bufB + (size_t)NN * DD;      // G*D
    float* counts   = pooled + GG * DD;            // G  (contiguous after pooled)
    float* zbuf     = counts + GG;                 // G*H

    const int T = 256;

    // degree + symmetric norm
    k_fill<<<cdiv(NN, T), T, 0, stream>>>(deg, 1.0f, NN);              // self loop
    k_deg_edges<<<cdiv(EE, T), T, 0, stream>>>(dst, deg, EE);
    k_dinv<<<cdiv(NN, T), T, 0, stream>>>(deg, dinv, selfn, NN);
    k_norm<<<cdiv(EE, T), T, 0, stream>>>(src, dst, dinv, norm, EE);

    // GCN layers
    const int n4 = NN * DD / 4;
    const int rowTiles = NN / 16;                 // 6250
    for (int l = 0; l < LL; ++l) {
        const float* hin = (l == 0) ? x : bufA;
        k_gemm<<<cdiv(rowTiles, 4), T, 0, stream>>>(hin, Ws + (size_t)l * DD * DD, bufB);
        k_selfinit<<<cdiv(n4, T), T, 0, stream>>>((const float4*)bufB, selfn,
                                                  (float4*)bufA, n4);
        k_scatter<<<cdiv((long long)EE * 32, T), T, 0, stream>>>(bufB, src, dst, norm,
                                                                 bufA, EE);
        k_bn_relu<<<cdiv((long long)NN * DD, T), T, 0, stream>>>(
            bufA, bs + l * DD, gam + l * DD, bet + l * DD, mu + l * DD, var + l * DD,
            NN * DD);
    }

    // global mean pool
    k_fill<<<cdiv(GG * DD + GG, T), T, 0, stream>>>(pooled, 0.0f, GG * DD + GG);
    k_counts<<<cdiv(NN, T), T, 0, stream>>>(batch, counts, NN);
    k_poolsum<<<cdiv((long long)NN * 32, T), T, 0, stream>>>(bufA, batch, pooled, NN);

    // MLP head
    k_fc1<<<cdiv(GG * HH, T), T, 0, stream>>>(pooled, counts, fc1w, fc1b, zbuf);
    k_fc2<<<cdiv(GG * CC, T), T, 0, stream>>>(zbuf, fc2w, fc2b, out);
}